// SemiCurrSinkhornKnopp_76914274337442
// MI455X (gfx1250) — compile-verified
//
#include <hip/hip_runtime.h>
#include <hip/hip_bf16.h>
#include <math.h>
#include <stdint.h>

// ---------------- problem constants (from the reference) ----------------
// EPSILON=0.1, GAMMA=1.0, RHO=1.0, SEMI_USE=True, 50 iters, stoperr=1e-6
// CLUSTER_SIZES sum to exactly 100 -> prior = sizes/100, Pb[7] = 1-RHO = 0
#define K_COLS        7
#define M_COLS        8            // k+1 (last column of Q is exp(0)=1)
#define NUM_ITERS     50
#define EPS_INV       10.0         // 1/EPSILON
#define FI_EXP        (1.0/1.1)    // GAMMA/(GAMMA+EPSILON)
#define STOPERR_D     1e-6

__constant__ double c_Pb[8] = {0.17, 0.14, 0.15, 0.12, 0.05, 0.13, 0.24, 0.0};

// ---------------- tiling for the streaming iteration kernel ----------------
// Q stored as f32 [n,8] = 32B/row (computed in f64, rounded once; all
// accumulation stays f64 -> ~6e-8 relative error, halves dominant traffic).
#define TPB             256
#define WAVES_PER_BLK   8            // wave32
#define TILE_ROWS       32           // rows staged per async stage (per wave)
#define TILES_PER_WAVE  4
#define ROWS_PER_WAVE   (TILE_ROWS * TILES_PER_WAVE)       // 128
#define ROWS_PER_BLOCK  (ROWS_PER_WAVE * WAVES_PER_BLK)    // 1024
#define LDS_ROW_FL      12           // 8 floats + 4 pad -> 48B stride, 16B aligned,
                                     // <=2-way bank conflicts on ds readback
#define BUF_FL          (TILE_ROWS * LDS_ROW_FL)           // 384 floats / buffer
#define NUM_PART        64           // partial-sum slots for Q^T a

// low 32 bits of a generic pointer into LDS == the LDS byte offset used by
// ds / async-to-LDS instructions
__device__ __forceinline__ uint32_t lds_byte_off(const void* p) {
    return (uint32_t)(uintptr_t)p;
}

// ---------------- kernel 0: init small state ----------------
__global__ void init_state_kernel(double* part, double* bvec, int* flag) {
    int t = threadIdx.x;
    for (int i = t; i < NUM_PART * 8; i += blockDim.x) part[i] = 0.0;
    if (t < 8)  bvec[t] = 0.125;        // b0 = 1/(k+1)
    if (t == 0) *flag   = 0;
}

// ---------------- kernel 1: Q = exp(log_softmax(P)/eps) (f64 math, f32 store) ----
__global__ __launch_bounds__(256) void build_q_kernel(
    const float* __restrict__ P, float* __restrict__ Q, int n) {
    int row = blockIdx.x * 256 + threadIdx.x;
    if (row >= n) return;
    const float* pr = P + (size_t)row * K_COLS;
    double p[K_COLS];
    #pragma unroll
    for (int j = 0; j < K_COLS; ++j) p[j] = (double)pr[j];
    double m = p[0];
    #pragma unroll
    for (int j = 1; j < K_COLS; ++j) m = fmax(m, p[j]);
    double s = 0.0;
    #pragma unroll
    for (int j = 0; j < K_COLS; ++j) s += exp(p[j] - m);
    double lse = m + log(s);
    float* qr = Q + (size_t)row * M_COLS;
    #pragma unroll
    for (int j = 0; j < K_COLS; ++j) qr[j] = (float)exp((p[j] - lse) * EPS_INV);
    qr[7] = 1.0f;
}

// ---------------- kernel 2: per-row Sinkhorn pass ----------------
// a_i = (1/n)/(Q_i . b);  partial[j] += sum_i Q_ij * a_i   (all math f64)
// Streams Q via per-wave double-buffered global_load_async_to_lds_b128:
// global side is contiguous 512B per instruction; LDS side lands in a
// 48B-padded row layout so each lane reads back its own row near-conflict-free.
__global__ __launch_bounds__(TPB) void sinkhorn_row_kernel(
    const float* __restrict__ Q, double* __restrict__ a,
    const double* __restrict__ bvec, double* __restrict__ part,
    const int* __restrict__ flag, int n) {
    if (*flag) return;                       // converged: iteration is a no-op

    __shared__ float lds[WAVES_PER_BLK * 2 * BUF_FL];   // 24 KiB

    const int tid  = threadIdx.x;
    const int wave = tid >> 5;
    const int lane = tid & 31;

    double b[8];
    #pragma unroll
    for (int j = 0; j < 8; ++j) b[j] = bvec[j];
    const double Pa = 1.0 / (double)n;

    float* wbuf = &lds[wave * 2 * BUF_FL];
    const uint32_t lds_base = lds_byte_off(wbuf);
    // 16B chunk c = lane + 32k: row = (lane>>1)+16k, half p = lane&1
    const uint32_t lds_lane = (uint32_t)((lane >> 1) * 48 + (lane & 1) * 16);
    const int      waveRow0 = blockIdx.x * ROWS_PER_BLOCK + wave * ROWS_PER_WAVE;

    double tacc[8];
    #pragma unroll
    for (int j = 0; j < 8; ++j) tacc[j] = 0.0;

    auto issue_tile = [&](int tile, int buf) {
        uint32_t g0 = (uint32_t)((waveRow0 + tile * TILE_ROWS) * 32) +
                      (uint32_t)(lane * 16);
        uint32_t l0 = lds_base + (uint32_t)(buf * (BUF_FL * 4)) + lds_lane;
        #pragma unroll
        for (int k = 0; k < 2; ++k) {
            asm volatile("global_load_async_to_lds_b128 %0, %1, %2"
                         :: "v"(l0 + (uint32_t)(k * 768)),   // 16 rows * 48B
                            "v"(g0 + (uint32_t)(k * 512)),
                            "s"(Q)
                         : "memory");
        }
    };

    issue_tile(0, 0);
    int buf = 0;
    for (int t = 0; t < TILES_PER_WAVE; ++t) {
        if (t + 1 < TILES_PER_WAVE) {
            issue_tile(t + 1, buf ^ 1);                       // prefetch next
            asm volatile("s_wait_asynccnt 0x2" ::: "memory"); // current tile done
        } else {
            asm volatile("s_wait_asynccnt 0x0" ::: "memory");
        }
        const float* rp = &wbuf[buf * BUF_FL + lane * LDS_ROW_FL];
        double q[8];
        #pragma unroll
        for (int j = 0; j < 8; ++j) q[j] = (double)rp[j];
        double denom = 0.0;
        #pragma unroll
        for (int j = 0; j < 8; ++j) denom += q[j] * b[j];
        double ai = Pa / denom;
        a[waveRow0 + t * TILE_ROWS + lane] = ai;
        #pragma unroll
        for (int j = 0; j < 8; ++j) tacc[j] += q[j] * ai;
        buf ^= 1;
    }

    // wave32 butterfly reduce of the 8 column partials
    #pragma unroll
    for (int j = 0; j < 8; ++j) {
        #pragma unroll
        for (int off = 16; off > 0; off >>= 1)
            tacc[j] += __shfl_xor(tacc[j], off, 32);
    }
    if (lane == 0) {
        double* slot = part + (size_t)(blockIdx.x & (NUM_PART - 1)) * 8;
        #pragma unroll
        for (int j = 0; j < 8; ++j) atomicAdd(&slot[j], tacc[j]);
    }
}

// ---------------- kernel 3: b-update + convergence test (1 wave) ----------------
__global__ void update_b_kernel(double* part, double* bvec, int* flag) {
    if (*flag) return;
    int j = threadIdx.x;                     // blockDim.x == 32 (single wave)
    double nb = 0.0, ob = 0.0;
    if (j < 8) {
        double s = 0.0;
        for (int p = 0; p < NUM_PART; ++p) s += part[p * 8 + j];
        ob = bvec[j];
        nb = c_Pb[j] / s;                    // j==7: Pb=0 -> b7=0
        if (j < 7) nb = pow(nb, FI_EXP);
        bvec[j] = nb;
    }
    // same wave => lanes 8..31 zero the partials only after lanes 0..7 read them
    for (int i = j; i < NUM_PART * 8; i += 32) part[i] = 0.0;
    double d = nb - ob; d *= d;
    #pragma unroll
    for (int off = 16; off > 0; off >>= 1) d += __shfl_xor(d, off, 32);
    if (j == 0 && sqrt(d) <= STOPERR_D) *flag = 1;
}

// ---------------- kernel 4: plan epilogue ----------------
// plan_ij = n * a_i * q_ij * b_j  (row-normalized; f64 math, f32 store)
__global__ __launch_bounds__(256) void plan_kernel(
    const float* __restrict__ Q, const double* __restrict__ a,
    const double* __restrict__ bvec, float* __restrict__ out, int n) {
    int row = blockIdx.x * 256 + threadIdx.x;
    if (row >= n) return;
    double bb[K_COLS];
    #pragma unroll
    for (int j = 0; j < K_COLS; ++j) bb[j] = bvec[j];
    const float* qr = Q + (size_t)row * M_COLS;
    double s = a[row] * (double)n;
    float* orow = out + (size_t)row * K_COLS;
    #pragma unroll
    for (int j = 0; j < K_COLS; ++j)
        orow[j] = (float)(s * (double)qr[j] * bb[j]);
}

// ---------------- launcher ----------------
extern "C" void kernel_launch(void* const* d_in, const int* in_sizes, int n_in,
                              void* d_out, int out_size, void* d_ws, size_t ws_size,
                              hipStream_t stream) {
    const float* P = (const float*)d_in[0];
    const int n = in_sizes[0] / K_COLS;          // 2,097,152 (multiple of 1024)

    // workspace layout (~80 MiB):
    //   Q: n*8 f32, a: n f64, part: 64*8 f64, bvec: 8 f64, flag: int
    char*   ws   = (char*)d_ws;
    float*  Q    = (float*)ws;
    double* a    = (double*)(ws + (size_t)n * 32);
    double* part = (double*)(ws + (size_t)n * 32 + (size_t)n * 8);
    double* bvec = part + NUM_PART * 8;
    int*    flag = (int*)(bvec + 8);

    init_state_kernel<<<1, 64, 0, stream>>>(part, bvec, flag);
    build_q_kernel<<<(n + 255) / 256, 256, 0, stream>>>(P, Q, n);

    const int nblk = n / ROWS_PER_BLOCK;         // 2048
    for (int it = 0; it < NUM_ITERS; ++it) {
        sinkhorn_row_kernel<<<nblk, TPB, 0, stream>>>(Q, a, bvec, part, flag, n);
        update_b_kernel<<<1, 32, 0, stream>>>(part, bvec, flag);
    }
    plan_kernel<<<(n + 255) / 256, 256, 0, stream>>>(Q, a, bvec, (float*)d_out, n);
}